// MSE_Penalty_19292993093700
// MI455X (gfx1250) — compile-verified
//
#include <hip/hip_runtime.h>

#define HW      (512 * 512)
#define W_DIM   512
#define NBATCH  256
#define BLK_PER_BATCH 64          // 64 blocks * 256 thr * 16 elems = 262144 = HW
#define NPART   (NBATCH * BLK_PER_BATCH)
#define AMAX_THREADS 1024

typedef __attribute__((ext_vector_type(2))) float v2f;
typedef __attribute__((ext_vector_type(4))) float v4f;
typedef __attribute__((ext_vector_type(8))) float v8f;

// ---------------------------------------------------------------------------
// Wave32 sum-reduction via V_WMMA_F32_16X16X4_F32.
// A (16x4 f32): lane L (0..15) holds row M=L, K=0 in a[0]; lane L+16 holds
// row M=L, K=2 in a[0] (per ISA 32-bit A layout). a[1] (K=1/K=3) = 0.
// B (4x16) = all ones  =>  D[m,n] = x(lane m) + x(lane m+16) = rowsum(m).
// Per-lane sum of the 8 D VGPRs gives rowsums 0..7 (lanes 0-15) and
// rowsums 8..15 (lanes 16-31); one shfl_xor(16) completes the 32-lane total.
// EXEC must be all ones at the call site (all callers are divergence-free).
// ---------------------------------------------------------------------------
__device__ __forceinline__ float wave_reduce_sum(float x) {
    v2f a; a[0] = x;    a[1] = 0.0f;
    v2f b; b[0] = 1.0f; b[1] = 1.0f;
    v8f c = {};
    c = __builtin_amdgcn_wmma_f32_16x16x4_f32(
            /*neg_a=*/false, a, /*neg_b=*/false, b,
            /*c_mod=*/(short)0, c, /*reuse_a=*/false, /*reuse_b=*/false);
    float s = c[0] + c[1] + c[2] + c[3] + c[4] + c[5] + c[6] + c[7];
    s += __shfl_xor(s, 16, 32);
    return s;   // every lane holds the full 32-lane sum
}

// ---------------------------------------------------------------------------
// Kernel 1: per-batch (first) argmax + max value of target. One block/batch,
// 1024 threads (32 waves) so the memory system sees plenty of in-flight b128
// loads. target stays regular-temporal: the MSE pass re-reads it and the
// 192 MB L2 can retain most of the 256 MB between passes.
// ---------------------------------------------------------------------------
__global__ void k_argmax(const float* __restrict__ target,
                         float* __restrict__ mval, int* __restrict__ mpos) {
    const int b   = blockIdx.x;
    const int tid = threadIdx.x;                     // AMAX_THREADS threads
    const v4f* t4 = (const v4f*)(target + (size_t)b * HW);

    float bestV = -__builtin_inff();
    int   bestI = 0;
    // Indices visited by a thread are strictly increasing, so strict '>'
    // keeps the first (lowest-index) maximum within a thread.
    for (int i = 0; i < HW / 4 / AMAX_THREADS; ++i) {
        const int q  = i * AMAX_THREADS + tid;       // coalesced
        const v4f v  = t4[q];
        const int e0 = q * 4;
        if (v[0] > bestV) { bestV = v[0]; bestI = e0;     }
        if (v[1] > bestV) { bestV = v[1]; bestI = e0 + 1; }
        if (v[2] > bestV) { bestV = v[2]; bestI = e0 + 2; }
        if (v[3] > bestV) { bestV = v[3]; bestI = e0 + 3; }
    }

    __shared__ float sv[AMAX_THREADS];
    __shared__ int   si[AMAX_THREADS];
    sv[tid] = bestV; si[tid] = bestI;
    __syncthreads();
    for (int off = AMAX_THREADS / 2; off > 0; off >>= 1) {
        if (tid < off) {
            const float v2 = sv[tid + off];
            const int   i2 = si[tid + off];
            if (v2 > sv[tid] || (v2 == sv[tid] && i2 < si[tid])) {
                sv[tid] = v2; si[tid] = i2;
            }
        }
        __syncthreads();
    }
    if (tid == 0) { mval[b] = sv[0]; mpos[b] = si[0]; }
}

// ---------------------------------------------------------------------------
// Kernel 2: penalty-weighted squared-error partial sums.
// grid = (BLK_PER_BATCH, NBATCH), 256 threads. Per-batch params are uniform
// within a block (blockIdx.y) -> scalar loads / SALU math.
// `outp` is read exactly once across the whole problem -> non-temporal loads
// so it doesn't evict `target` lines from the 192 MB L2.
// ---------------------------------------------------------------------------
__global__ void k_mse(const float* __restrict__ outp, const float* __restrict__ tgt,
                      const float* __restrict__ mval, const int* __restrict__ mpos,
                      float* __restrict__ partial) {
    const int b   = blockIdx.y;
    const int bx  = blockIdx.x;
    const int tid = threadIdx.x;                     // 256 threads
    const size_t base = (size_t)b * HW;

    const float mv = mval[b];
    const int   mp = mpos[b];
    const int ph = mp >> 9, pw = mp & 511;
    const int top  = max(ph - 32, 0), bot   = min(ph + 32, 512);
    const int left = max(pw - 32, 0), right = min(pw + 32, 512);
    const int Lh = bot - top,  Lw = right - left;
    const float invH = 9.0f / (float)max((Lh + 1) / 2 - 1, 1);
    const float invW = 9.0f / (float)max((Lw + 1) / 2 - 1, 1);
    const bool active = (mv >= 0.5f);

    float acc = 0.0f;
#pragma unroll
    for (int i = 0; i < 4; ++i) {
        const int e = bx * 4096 + (i * 256 + tid) * 4;   // element in batch
        const v4f o = __builtin_nontemporal_load((const v4f*)(outp + base + e));
        const v4f t = *(const v4f*)(tgt + base + e);

        const int h  = e >> 9;
        const int w0 = e & 511;                          // 4 elems, same row
        const int jh = h - top;
        const bool rowin = active & (jh >= 0) & (jh < Lh);
        const float vh = 1.0f + (float)min(jh, Lh - 1 - jh) * invH;

        float d, pen;
        int jw;

        jw = w0 - left;
        pen = (rowin & (jw >= 0) & (jw < Lw))
                ? vh * (1.0f + (float)min(jw, Lw - 1 - jw) * invW) : 1.0f;
        d = o[0] - t[0]; acc = fmaf(d * d, pen, acc);

        jw = w0 + 1 - left;
        pen = (rowin & (jw >= 0) & (jw < Lw))
                ? vh * (1.0f + (float)min(jw, Lw - 1 - jw) * invW) : 1.0f;
        d = o[1] - t[1]; acc = fmaf(d * d, pen, acc);

        jw = w0 + 2 - left;
        pen = (rowin & (jw >= 0) & (jw < Lw))
                ? vh * (1.0f + (float)min(jw, Lw - 1 - jw) * invW) : 1.0f;
        d = o[2] - t[2]; acc = fmaf(d * d, pen, acc);

        jw = w0 + 3 - left;
        pen = (rowin & (jw >= 0) & (jw < Lw))
                ? vh * (1.0f + (float)min(jw, Lw - 1 - jw) * invW) : 1.0f;
        d = o[3] - t[3]; acc = fmaf(d * d, pen, acc);
    }

    // Wave-level reduction on the WMMA path (EXEC all ones here).
    const float wsum = wave_reduce_sum(acc);

    __shared__ float wpart[8];
    const int wid  = tid >> 5;
    const int lane = tid & 31;
    if (lane == 0) wpart[wid] = wsum;
    __syncthreads();
    if (tid == 0) {
        float s = 0.0f;
#pragma unroll
        for (int i = 0; i < 8; ++i) s += wpart[i];
        partial[b * BLK_PER_BATCH + bx] = s;
    }
}

// ---------------------------------------------------------------------------
// Kernel 3: deterministic final reduction of 16384 partials (double accum).
// ---------------------------------------------------------------------------
__global__ void k_final(const float* __restrict__ partial, float* __restrict__ res) {
    const int tid = threadIdx.x;                     // 256 threads
    double s = 0.0;
    for (int i = tid; i < NPART; i += 256) s += (double)partial[i];
    __shared__ double sd[256];
    sd[tid] = s;
    __syncthreads();
    for (int off = 128; off > 0; off >>= 1) {
        if (tid < off) sd[tid] += sd[tid + off];
        __syncthreads();
    }
    if (tid == 0) res[0] = (float)(sd[0] / (double)((size_t)NBATCH * HW));
}

// ---------------------------------------------------------------------------
extern "C" void kernel_launch(void* const* d_in, const int* in_sizes, int n_in,
                              void* d_out, int out_size, void* d_ws, size_t ws_size,
                              hipStream_t stream) {
    const float* output = (const float*)d_in[0];
    const float* target = (const float*)d_in[1];

    float* mval    = (float*)d_ws;                            // 256 f32
    int*   mpos    = (int*)((char*)d_ws + NBATCH * 4);        // 256 i32
    float* partial = (float*)((char*)d_ws + 2 * NBATCH * 4);  // 16384 f32

    k_argmax<<<NBATCH, AMAX_THREADS, 0, stream>>>(target, mval, mpos);

    dim3 g2(BLK_PER_BATCH, NBATCH);
    k_mse<<<g2, 256, 0, stream>>>(output, target, mval, mpos, partial);

    k_final<<<1, 256, 0, stream>>>(partial, (float*)d_out);
}